// EntityEmbedding_67568425501223
// MI455X (gfx1250) — compile-verified
//
#include <hip/hip_runtime.h>
#include <stdint.h>

// Problem constants (from the reference)
#define VOCAB   1000000
#define HID     64          // hidden size (floats per embedding row; 256 B/row)
#define POS     50          // positions pooled per batch row
#define BATCH   16384
#define RPG     8           // rows per TDM gather (32-bit index mode => max 8)
#define NG      ((POS + RPG - 1) / RPG)   // 7 gather groups (8,8,8,8,8,8,2)
#define DEPTH   4           // LDS ring buffers per wave (pipeline depth)
#define WAVES   8           // waves per block (wave32 => 256 threads)
#define THREADS (WAVES * 32)
#define BUFBYTES (RPG * HID * 4)          // 2 KB per gather group

#if defined(__has_builtin)
# if __has_builtin(__builtin_amdgcn_tensor_load_to_lds)
#  define HAVE_TDM 1
# endif
#endif

typedef unsigned int v4u __attribute__((ext_vector_type(4)));
typedef int          v4i __attribute__((ext_vector_type(4)));
typedef int          v8i __attribute__((ext_vector_type(8)));

// LDS (addrspace 3) pointer for direct ds_load_b64; volatile because the TDM
// engine writes LDS behind the compiler's back.
typedef volatile const __attribute__((address_space(3))) unsigned long long* lds_u64p;

template <int N>
__device__ __forceinline__ void wait_tensorcnt() {
#if defined(__has_builtin) && __has_builtin(__builtin_amdgcn_s_wait_tensorcnt)
  __builtin_amdgcn_s_wait_tensorcnt(N);
#else
  asm volatile("s_wait_tensorcnt %0" :: "i"(N) : "memory");
#endif
}

#ifdef HAVE_TDM
// Issue one TDM gather of cnt (<=8) embedding rows into LDS at byte offset
// lds_byte. Descriptor per CDNA5 ISA:
//   group0: count=1 | gather_index_size=1(32b)<<30 | gather_mode=1<<31 ;
//           lds_addr ; global_addr[31:0] ; global_addr[56:32] | type(2)<<30
//   group1: data_size=2(4B)<<16 ; tensor_dim0=64 (bits 79:48) ;
//           tensor_dim1=VOCAB (bits 111:80) ; tile_dim0=64 (bits 127:112) ;
//           tile_dim1=cnt (#valid indices) ; tensor_dim0_stride=64
//   group2/3: row indices (32-bit index mode: 4 + 4)
template <int G>
__device__ __forceinline__ void tdm_gather8(unsigned lds_byte,
                                            unsigned long long wbase,
                                            const int* __restrict__ xr) {
  constexpr int base = G * RPG;
  constexpr int cnt  = (base + RPG <= POS) ? RPG : (POS - base);
  int id[RPG];
#pragma unroll
  for (int r = 0; r < RPG; ++r) {
    // wave-uniform index (one batch row per wave) -> force SGPR for the D#
    id[r] = (r < cnt) ? __builtin_amdgcn_readfirstlane(xr[base + r]) : 0;
  }
  v4u g0 = {
    0xC0000001u,                                   // count=1, 32b idx, gather
    lds_byte,                                      // LDS dest (bytes)
    (unsigned)(wbase & 0xFFFFFFFFull),             // global_addr[31:0]
    (((unsigned)(wbase >> 32)) & 0x01FFFFFFu) | (2u << 30)  // [56:32] | type=2
  };
  v8i g1 = {
    (int)(2u << 16),                               // data_size = 4 bytes
    (int)(((unsigned)HID & 0xFFFFu) << 16),        // tensor_dim0 low16
    (int)((((unsigned)HID >> 16) & 0xFFFFu) |
          (((unsigned)VOCAB & 0xFFFFu) << 16)),    // dim0 hi16 | dim1 low16
    (int)((((unsigned)VOCAB >> 16) & 0xFFFFu) |
          (((unsigned)HID) << 16)),                // dim1 hi16 | tile_dim0
    (int)cnt,                                      // tile_dim1 = #valid idx
    (int)HID,                                      // tensor_dim0_stride low32
    0, 0                                           // stride hi / dim1_stride
  };
  v4i g2 = { id[0], id[1], id[2], id[3] };
  v4i g3 = { id[4], id[5], id[6], id[7] };
#if __has_include(<hip/amd_detail/amd_gfx1250_TDM.h>)
  // amdgpu-toolchain (clang-23 / therock-10.0): 6-arg form
  v8i g23 = { id[0], id[1], id[2], id[3], id[4], id[5], id[6], id[7] };
  __builtin_amdgcn_tensor_load_to_lds(g0, g1, g2, g3, g23, 0);
#else
  // ROCm 7.2 (clang-22): 5-arg form
  __builtin_amdgcn_tensor_load_to_lds(g0, g1, g2, g3, 0);
#endif
}

// One software-pipeline step: wait until ring slot G's gather is complete,
// reduce its (<=8) rows from LDS via ds_load_b64, then refill the slot with
// gather group G+DEPTH.
template <int G>
__device__ __forceinline__ void pipeline_step(unsigned waveByte,
                                              unsigned long long wbase,
                                              const int* __restrict__ xr,
                                              int lane, float& ax, float& ay) {
  constexpr int issued = (G + DEPTH < NG) ? (G + DEPTH) : NG;
  wait_tensorcnt<issued - (G + 1)>();              // slot G's data has landed
  constexpr int cnt = (G * RPG + RPG <= POS) ? RPG : (POS - G * RPG);
  const unsigned slotByte = waveByte + (unsigned)((G & (DEPTH - 1)) * BUFBYTES);
  const unsigned laneByte = slotByte + (unsigned)(lane * 8);
#pragma unroll
  for (int rr = 0; rr < cnt; ++rr) {
    const unsigned long long q =
        *(lds_u64p)(uintptr_t)(laneByte + (unsigned)(rr * HID * 4));
    ax += __uint_as_float((unsigned)(q & 0xFFFFFFFFull));
    ay += __uint_as_float((unsigned)(q >> 32));
  }
  if constexpr (G + DEPTH < NG) {
    tdm_gather8<G + DEPTH>(slotByte, wbase, xr);   // refill same ring slot
  }
}
#endif // HAVE_TDM

// One wave per batch row; lane L owns hidden elements 2L, 2L+1.
// Each embedding row = 256 B = one fully coalesced transfer.
__global__ __launch_bounds__(THREADS)
void EntityEmbedding_gather_mean(const int* __restrict__ x,
                                 const float* __restrict__ w,
                                 float* __restrict__ out) {
  const int lane = threadIdx.x & 31;
  const int wave = threadIdx.x >> 5;
  const int row  = blockIdx.x * WAVES + wave;       // grid sized exactly
  const int* xr  = x + (size_t)row * POS;

  float ax = 0.0f, ay = 0.0f;

#ifdef HAVE_TDM
  // 4-deep TDM gather ring: DEPTH x 8 rows x 256 B per wave (8 KB),
  // 64 KB per 256-thread block (LDS budget: 320 KB/WGP => ~5 blocks/WGP).
  __shared__ float smem[WAVES * DEPTH * RPG * HID];
  const unsigned smemBase = (unsigned)(size_t)(&smem[0]);   // LDS byte offset
  const unsigned waveByte = smemBase + (unsigned)(wave * DEPTH * BUFBYTES);
  const unsigned long long wbase = (unsigned long long)(uintptr_t)w;

  // Prime the pipeline: 4 gathers (8 KB of row data) in flight per wave.
  tdm_gather8<0>(waveByte + 0 * BUFBYTES, wbase, xr);
  tdm_gather8<1>(waveByte + 1 * BUFBYTES, wbase, xr);
  tdm_gather8<2>(waveByte + 2 * BUFBYTES, wbase, xr);
  tdm_gather8<3>(waveByte + 3 * BUFBYTES, wbase, xr);

  pipeline_step<0>(waveByte, wbase, xr, lane, ax, ay);  // wait<=3, issue g4
  pipeline_step<1>(waveByte, wbase, xr, lane, ax, ay);  // wait<=3, issue g5
  pipeline_step<2>(waveByte, wbase, xr, lane, ax, ay);  // wait<=3, issue g6
  pipeline_step<3>(waveByte, wbase, xr, lane, ax, ay);  // wait<=3
  pipeline_step<4>(waveByte, wbase, xr, lane, ax, ay);  // wait<=2
  pipeline_step<5>(waveByte, wbase, xr, lane, ax, ay);  // wait<=1
  pipeline_step<6>(waveByte, wbase, xr, lane, ax, ay);  // wait<=0 (2 rows)
#else
  // Fallback: register-resident gather. 50 independent global_load_b64 per
  // lane in flight (fully unrolled), coalesced 256 B per row per wave.
#pragma unroll
  for (int p = 0; p < POS; ++p) {
    const int idx = xr[p];
    const float2 v = *(const float2*)(w + (size_t)idx * HID + lane * 2);
    ax += v.x;
    ay += v.y;
  }
#endif

  const float sc = 1.0f / (float)POS;
  float2 r;
  r.x = ax * sc;
  r.y = ay * sc;
  *(float2*)(out + (size_t)row * HID + lane * 2) = r;
}

extern "C" void kernel_launch(void* const* d_in, const int* in_sizes, int n_in,
                              void* d_out, int out_size, void* d_ws, size_t ws_size,
                              hipStream_t stream) {
  (void)in_sizes; (void)n_in; (void)out_size; (void)d_ws; (void)ws_size;
  const int*   x = (const int*)d_in[0];      // [BATCH, POS] int32
  const float* w = (const float*)d_in[1];    // [VOCAB, HID] fp32
  float* out = (float*)d_out;                // [BATCH, HID] fp32

  dim3 grid(BATCH / WAVES);                  // 2048 blocks
  dim3 block(THREADS);                       // 256 threads = 8 wave32
  EntityEmbedding_gather_mean<<<grid, block, 0, stream>>>(x, w, out);
}